// BEVFormerEncoderLayer_58317065945169
// MI455X (gfx1250) — compile-verified
//
#include <hip/hip_runtime.h>
#include <hip/hip_bf16.h>
#include <math.h>

// ---------------- problem constants (fixed by the reference module) ---------
#define QN    2500      // BEV_H * BEV_W
#define CE    256       // EMBED
#define NH    8         // HEADS
#define HD    32        // HEAD_DIM
#define FFD   512       // FF
#define NCAM  6
#define BEVH  50
#define BEVW  50
#define LTOT  19560     // sum of level sizes

typedef __attribute__((ext_vector_type(2))) float v2f;
typedef __attribute__((ext_vector_type(8))) float v8f;
typedef __attribute__((ext_vector_type(4))) unsigned int u32x4;
typedef __attribute__((ext_vector_type(8))) int i32x8;
typedef __attribute__((ext_vector_type(4))) int i32x4;

// ---------------------------------------------------------------------------
// Generic WMMA f32 GEMM: out[M,N] = A[M,K] @ W[N,K]^T + bias (+resid) (ReLU?)
// One wave -> 16M x 64N tile (4 accumulators reuse each A fragment 4x).
// K stepped by 8: all 10 loads are issued into locals BEFORE the 8 WMMAs so
// the scheduler can clause them and use partial s_wait_loadcnt values.
// A-frag: lanes 0-15 hold M=0..15 K={k,k+1}; lanes 16-31 same M, K={k+2,k+3}.
// B-frag: same pattern over the N rows of W (row-major N x K).
// C/D: lane covers N=lane&15; vgpr r covers M = r + 8*(lane>>4).
// ---------------------------------------------------------------------------
__global__ __launch_bounds__(128) void gemm_wmma_bias(
    const float* __restrict__ A, const float* __restrict__ W,
    const float* __restrict__ bias, const float* __restrict__ resid,
    float* __restrict__ out, int M, int N, int K, int relu)
{
    const int lane = threadIdx.x & 31;
    const int wv   = threadIdx.x >> 5;
    const int ntiles = (N + 63) >> 6;
    const int mtiles = (M + 15) >> 4;
    const int t = blockIdx.x * 4 + wv;
    if (t >= mtiles * ntiles) return;           // wave-uniform: EXEC stays all-1s
    const int mt = t / ntiles, nt = t % ntiles;
    const int m0 = mt * 16, n0 = nt * 64;
    const int half = lane >> 4;
    const int sub  = lane & 15;

    int ma = m0 + sub; if (ma > M - 1) ma = M - 1;         // clamp tail rows
    const float* __restrict__ arow = A + (size_t)ma * K;
    const float* __restrict__ wrow = W + (size_t)(n0 + sub) * K;
    const int koff = half * 2;

    v8f acc[4];
#pragma unroll
    for (int j = 0; j < 4; ++j) {
        float bv = bias[n0 + j * 16 + sub];
#pragma unroll
        for (int r = 0; r < 8; ++r) acc[j][r] = bv;
    }

    for (int k = 0; k < K; k += 8) {
        // ---- batch all loads for this K-step ----
        v2f a0 = *(const v2f*)(arow + k + koff);
        v2f a1 = *(const v2f*)(arow + k + 4 + koff);
        v2f b0[4], b1[4];
#pragma unroll
        for (int j = 0; j < 4; ++j) {
            const float* wr = wrow + (size_t)j * 16 * K + k + koff;
            b0[j] = *(const v2f*)(wr);
            b1[j] = *(const v2f*)(wr + 4);
        }
        // ---- 8 WMMAs consume them ----
#pragma unroll
        for (int j = 0; j < 4; ++j)
            acc[j] = __builtin_amdgcn_wmma_f32_16x16x4_f32(false, a0, false, b0[j],
                                                           (short)0, acc[j],
                                                           false, false);
#pragma unroll
        for (int j = 0; j < 4; ++j)
            acc[j] = __builtin_amdgcn_wmma_f32_16x16x4_f32(false, a1, false, b1[j],
                                                           (short)0, acc[j],
                                                           false, false);
    }

#pragma unroll
    for (int j = 0; j < 4; ++j) {
        const int nc = n0 + j * 16 + sub;
#pragma unroll
        for (int r = 0; r < 8; ++r) {
            int mo = m0 + half * 8 + r;
            if (mo < M) {
                float v = acc[j][r];
                if (resid) v += resid[(size_t)mo * N + nc];
                if (relu)  v = fmaxf(v, 0.f);
                out[(size_t)mo * N + nc] = v;
            }
        }
    }
}

// ---------------------------------------------------------------------------
// TDM-staged GEMM for the big SCA value projection: A[M,256] @ W[256,256]^T.
// The Tensor Data Mover DMAs K-chunks of W into LDS (double-buffered),
// padded 2 DWORDs every 32 so 16-row B-fragment ds reads are conflict-free.
// Block = 8 waves = 2 M-tiles x 4 N-chunks (covers 32 rows x 256 cols).
// ---------------------------------------------------------------------------
#define TKC     32                   // K-chunk (floats)
#define TROW    34                   // padded LDS row stride (floats)
#define TBUF    (CE * TROW)          // one buffer: 256 rows * 34 floats

__device__ __forceinline__ void tdm_load_w_chunk(const float* gsrc,
                                                 unsigned lds_byte_off)
{
    unsigned long long ga = (unsigned long long)(size_t)gsrc;
    u32x4 g0;
    g0[0] = 1u;                                         // count=1 (valid), user mode
    g0[1] = lds_byte_off;                               // lds_addr
    g0[2] = (unsigned)(ga & 0xffffffffu);               // global_addr[31:0]
    g0[3] = (unsigned)((ga >> 32) & 0x1ffffffu)         // global_addr[56:32]
          | (2u << 30);                                 // type = 2 ("image")
    i32x8 g1;
    g1[0] = (int)((2u << 16)                            // data_size = 4B
          | (1u << 20)                                  // pad_enable
          | (4u << 22)                                  // pad_interval: 32 DWORDs
          | (1u << 25));                                // pad_amount: 2 DWORDs
    g1[1] = (int)((unsigned)(CE & 0xffff) << 16);       // tensor_dim0 lo (K=256)
    g1[2] = (int)(((unsigned)CE >> 16)                  // tensor_dim0 hi
          | ((unsigned)(CE & 0xffff) << 16));           // tensor_dim1 lo (N=256)
    g1[3] = (int)(((unsigned)CE >> 16)                  // tensor_dim1 hi
          | ((unsigned)TKC << 16));                     // tile_dim0 = 32
    g1[4] = (int)(unsigned)CE;                          // tile_dim1 = 256, tile_dim2 = 0
    g1[5] = CE;                                         // tensor_dim0_stride lo = 256
    g1[6] = 0;                                          // stride hi / dim1_stride lo
    g1[7] = 0;
    i32x4 gz4; gz4[0] = 0; gz4[1] = 0; gz4[2] = 0; gz4[3] = 0;
    i32x8 gz8;
#pragma unroll
    for (int i = 0; i < 8; ++i) gz8[i] = 0;
    // 6-arg form (this toolchain): (g0, g1, g2, g3, g4, cpol)
    __builtin_amdgcn_tensor_load_to_lds(g0, g1, gz4, gz4, gz8, 0);
}

__global__ __launch_bounds__(256) void gemm_wmma_tdm(
    const float* __restrict__ A, const float* __restrict__ W,
    const float* __restrict__ bias, float* __restrict__ out, int M)
{
    extern __shared__ float lds[];                      // 2 * TBUF floats
    const int lane = threadIdx.x & 31;
    const int wv   = threadIdx.x >> 5;                  // 0..7
    const int half = lane >> 4;
    const int sub  = lane & 15;
    const int mt   = blockIdx.x * 2 + (wv >> 2);
    const int n0   = (wv & 3) * 64;
    const int m0   = mt * 16;
    const int koff = half * 2;

    int ma = m0 + sub; if (ma > M - 1) ma = M - 1;
    const float* __restrict__ arow = A + (size_t)ma * CE;

    v8f acc[4];
#pragma unroll
    for (int j = 0; j < 4; ++j) {
        float bv = bias[n0 + j * 16 + sub];
#pragma unroll
        for (int r = 0; r < 8; ++r) acc[j][r] = bv;
    }

    const int nchunks = CE / TKC;                       // 8
    if (wv == 0) tdm_load_w_chunk(W, 0);                // prologue -> buffer 0
    for (int c = 0; c < nchunks; ++c) {
        if (wv == 0) __builtin_amdgcn_s_wait_tensorcnt(0);
        __syncthreads();                                // LDS buf[c&1] ready
        if (wv == 0 && c + 1 < nchunks)                 // prefetch next chunk
            tdm_load_w_chunk(W + (c + 1) * TKC, (unsigned)(((c + 1) & 1) * TBUF * 4));

        const float* __restrict__ wb = lds + (c & 1) * TBUF;
        const int kg = c * TKC;
#pragma unroll
        for (int kk = 0; kk < TKC; kk += 8) {
            // batch loads: 2 global A fragments + 8 LDS B fragments
            v2f a0 = *(const v2f*)(arow + kg + kk + koff);
            v2f a1 = *(const v2f*)(arow + kg + kk + 4 + koff);
            v2f b0[4], b1[4];
#pragma unroll
            for (int j = 0; j < 4; ++j) {
                const float* wr = wb + (n0 + j * 16 + sub) * TROW + kk + koff;
                b0[j] = *(const v2f*)(wr);
                b1[j] = *(const v2f*)(wr + 4);
            }
#pragma unroll
            for (int j = 0; j < 4; ++j)
                acc[j] = __builtin_amdgcn_wmma_f32_16x16x4_f32(false, a0, false, b0[j],
                                                               (short)0, acc[j],
                                                               false, false);
#pragma unroll
            for (int j = 0; j < 4; ++j)
                acc[j] = __builtin_amdgcn_wmma_f32_16x16x4_f32(false, a1, false, b1[j],
                                                               (short)0, acc[j],
                                                               false, false);
        }
        __syncthreads();                                // reads done before reuse
    }

#pragma unroll
    for (int j = 0; j < 4; ++j) {
        const int nc = n0 + j * 16 + sub;
#pragma unroll
        for (int r = 0; r < 8; ++r) {
            int mo = m0 + half * 8 + r;
            if (mo < M) out[(size_t)mo * CE + nc] = acc[j][r];
        }
    }
}

// ------------------------------ elementwise --------------------------------
__global__ void add_kernel(const float* __restrict__ a, const float* __restrict__ b,
                           float* __restrict__ o, int n)
{
    int i = blockIdx.x * blockDim.x + threadIdx.x;
    if (i < n) o[i] = a[i] + b[i];
}

// q2 = concat(prev_bev[0], q) along channel (rows of 512)
__global__ void concat_kernel(const float* __restrict__ pb0, const float* __restrict__ qb,
                              float* __restrict__ q2, int n)
{
    int i = blockIdx.x * blockDim.x + threadIdx.x;
    if (i >= n) return;
    int r = i >> 9, c = i & 511;
    q2[i] = (c < CE) ? pb0[r * CE + c] : qb[r * CE + (c - CE)];
}

// mean of two temporal copies: o[i] = 0.5*(ms[i] + ms[QN*CE + i])
__global__ void mean2_kernel(const float* __restrict__ ms, float* __restrict__ o, int n)
{
    int i = blockIdx.x * blockDim.x + threadIdx.x;
    if (i < n) o[i] = 0.5f * (ms[i] + ms[(size_t)n + i]);
}

// softmax over contiguous groups of 4 (TSA attention weights)
__global__ void softmax4_kernel(float* __restrict__ x, int groups)
{
    int g = blockIdx.x * blockDim.x + threadIdx.x;
    if (g >= groups) return;
    float* p = x + (size_t)g * 4;
    float a0 = p[0], a1 = p[1], a2 = p[2], a3 = p[3];
    float m = fmaxf(fmaxf(a0, a1), fmaxf(a2, a3));
    float e0 = __expf(a0 - m), e1 = __expf(a1 - m);
    float e2 = __expf(a2 - m), e3 = __expf(a3 - m);
    float inv = 1.0f / (e0 + e1 + e2 + e3);
    p[0] = e0 * inv; p[1] = e1 * inv; p[2] = e2 * inv; p[3] = e3 * inv;
}

// softmax over contiguous groups of 32 (SCA attention weights)
__global__ void softmax32_kernel(float* __restrict__ x, int groups)
{
    int g = blockIdx.x * blockDim.x + threadIdx.x;
    if (g >= groups) return;
    float* p = x + (size_t)g * 32;
    float v[32];
    float m = -3.402823e38f;
#pragma unroll
    for (int i = 0; i < 32; ++i) { v[i] = p[i]; m = fmaxf(m, v[i]); }
    float s = 0.f;
#pragma unroll
    for (int i = 0; i < 32; ++i) { v[i] = __expf(v[i] - m); s += v[i]; }
    float inv = 1.0f / s;
#pragma unroll
    for (int i = 0; i < 32; ++i) p[i] = v[i] * inv;
}

// ----------------------------- layer norm ----------------------------------
__global__ __launch_bounds__(32) void layernorm_kernel(
    const float* __restrict__ x, const float* __restrict__ g,
    const float* __restrict__ b, float* __restrict__ y)
{
    const int row = blockIdx.x;
    const int lane = threadIdx.x;
    const float* xr = x + (size_t)row * CE;
    float v[8];
    float s = 0.f;
#pragma unroll
    for (int i = 0; i < 8; ++i) { v[i] = xr[lane + i * 32]; s += v[i]; }
#pragma unroll
    for (int m = 16; m >= 1; m >>= 1) s += __shfl_xor(s, m, 32);
    const float mu = s * (1.0f / CE);
    float var = 0.f;
#pragma unroll
    for (int i = 0; i < 8; ++i) { float d = v[i] - mu; var += d * d; }
#pragma unroll
    for (int m = 16; m >= 1; m >>= 1) var += __shfl_xor(var, m, 32);
    var *= (1.0f / CE);
    const float inv = rsqrtf(var + 1e-5f);
#pragma unroll
    for (int i = 0; i < 8; ++i) {
        int c = lane + i * 32;
        y[(size_t)row * CE + c] = (v[i] - mu) * inv * g[c] + b[c];
    }
}

// --------------------------- deformable sampling ---------------------------
__device__ __forceinline__ float bilin_tap(const float* __restrict__ img,
                                           int x, int y, int W, int H, float wt)
{
    if ((unsigned)x < (unsigned)W && (unsigned)y < (unsigned)H)
        return wt * img[(size_t)(y * W + x) * CE];
    return 0.f;
}

// TSA msda: one wave per (b2, q, h); lane = head-dim channel.
__global__ __launch_bounds__(256) void tsa_msda_kernel(
    const float* __restrict__ vproj,   // (2, QN, 256)
    const float* __restrict__ off,     // (QN, 128)  [h*16 + b2*8 + p*2 + xy]
    const float* __restrict__ aw,      // (QN, 64)   [h*8 + b2*4 + p], softmaxed
    const float* __restrict__ ref2d,   // (2, QN, 1, 2)
    float* __restrict__ outms)         // (2, QN, 256)
{
    int wid = blockIdx.x * 8 + (threadIdx.x >> 5);
    int lane = threadIdx.x & 31;
    if (wid >= 2 * QN * NH) return;
    int h = wid % NH; int t = wid / NH;
    int q = t % QN;   int b2 = t / QN;

    const float rx = ref2d[((size_t)b2 * QN + q) * 2 + 0];
    const float ry = ref2d[((size_t)b2 * QN + q) * 2 + 1];
    const float* img = vproj + ((size_t)b2 * QN) * CE + h * HD + lane;
    const float* offq = off + (size_t)q * 128 + h * 16 + b2 * 8;
    const float* awq  = aw  + (size_t)q * 64  + h * 8  + b2 * 4;

    float accum = 0.f;
#pragma unroll
    for (int p = 0; p < 4; ++p) {
        float lx = rx + offq[p * 2 + 0] * (1.0f / BEVW);
        float ly = ry + offq[p * 2 + 1] * (1.0f / BEVH);
        float w  = awq[p];
        float x = lx * BEVW - 0.5f;
        float y = ly * BEVH - 0.5f;
        float xf = floorf(x), yf = floorf(y);
        float fx = x - xf, fy = y - yf;
        int x0 = (int)xf, y0 = (int)yf;
        float s = bilin_tap(img, x0,     y0,     BEVW, BEVH, (1.f - fx) * (1.f - fy))
                + bilin_tap(img, x0 + 1, y0,     BEVW, BEVH, fx * (1.f - fy))
                + bilin_tap(img, x0,     y0 + 1, BEVW, BEVH, (1.f - fx) * fy)
                + bilin_tap(img, x0 + 1, y0 + 1, BEVW, BEVH, fx * fy);
        accum += w * s;
    }
    outms[((size_t)b2 * QN + q) * CE + h * HD + lane] = accum;
}

// SCA msda: one wave per (cam, q, h); lane = head-dim channel; 4 levels x 8 pts.
__global__ __launch_bounds__(256) void sca_msda_kernel(
    const float* __restrict__ vproj,   // (6, LTOT, 256)
    const float* __restrict__ off,     // (QN, 512) [h*64 + lvl*16 + p*2 + xy]
    const float* __restrict__ aw,      // (QN, 256) [h*32 + lvl*8 + p], softmaxed
    const float* __restrict__ refimg,  // (6, 1, QN, 4, 2)
    float* __restrict__ osca)          // (6, QN, 256)
{
    const int Hs[4] = {92, 46, 23, 12};
    const int Ws[4] = {160, 80, 40, 20};
    const int st[4] = {0, 14720, 18400, 19320};

    int wid = blockIdx.x * 8 + (threadIdx.x >> 5);
    int lane = threadIdx.x & 31;
    if (wid >= NCAM * QN * NH) return;
    int h = wid % NH; int t = wid / NH;
    int q = t % QN;   int cam = t / QN;

    const float* refq = refimg + ((size_t)cam * QN + q) * 8;   // 4 z-anchors x 2
    const float* offq = off + (size_t)q * 512 + h * 64;
    const float* awq  = aw  + (size_t)q * 256 + h * 32;
    const float* base = vproj + ((size_t)cam * LTOT) * CE + h * HD + lane;

    float accum = 0.f;
#pragma unroll
    for (int lvl = 0; lvl < 4; ++lvl) {
        const int Wl = Ws[lvl], Hl = Hs[lvl];
        const float* img = base + (size_t)st[lvl] * CE;
        const float invw = 1.0f / Wl, invh = 1.0f / Hl;
#pragma unroll
        for (int p = 0; p < 8; ++p) {
            int z = p & 3;                                 // p = pp*4 + z
            float lx = refq[z * 2 + 0] + offq[lvl * 16 + p * 2 + 0] * invw;
            float ly = refq[z * 2 + 1] + offq[lvl * 16 + p * 2 + 1] * invh;
            float w  = awq[lvl * 8 + p];
            float x = lx * Wl - 0.5f;
            float y = ly * Hl - 0.5f;
            float xf = floorf(x), yf = floorf(y);
            float fx = x - xf, fy = y - yf;
            int x0 = (int)xf, y0 = (int)yf;
            float s = bilin_tap(img, x0,     y0,     Wl, Hl, (1.f - fx) * (1.f - fy))
                    + bilin_tap(img, x0 + 1, y0,     Wl, Hl, fx * (1.f - fy))
                    + bilin_tap(img, x0,     y0 + 1, Wl, Hl, (1.f - fx) * fy)
                    + bilin_tap(img, x0 + 1, y0 + 1, Wl, Hl, fx * fy);
            accum += w * s;
        }
    }
    osca[((size_t)cam * QN + q) * CE + h * HD + lane] = accum;
}

// ----------------------- camera hit mask + masked mean ---------------------
__global__ void hit_kernel(const unsigned char* __restrict__ mask,
                           float* __restrict__ hitf, float* __restrict__ cnt)
{
    int q = blockIdx.x * blockDim.x + threadIdx.x;
    if (q >= QN) return;
    float c = 0.f;
#pragma unroll
    for (int cam = 0; cam < NCAM; ++cam) {
        int s = 0;
#pragma unroll
        for (int z = 0; z < 4; ++z) s += mask[(size_t)cam * QN * 4 + q * 4 + z];
        float h = (s > 0) ? 1.f : 0.f;
        hitf[cam * QN + q] = h;
        c += h;
    }
    cnt[q] = fmaxf(c, 1.f);
}

__global__ void slots_kernel(const float* __restrict__ osca,
                             const float* __restrict__ hitf,
                             const float* __restrict__ cnt,
                             float* __restrict__ slots, int n)
{
    int i = blockIdx.x * blockDim.x + threadIdx.x;
    if (i >= n) return;
    int q = i >> 8;
    float s = 0.f;
#pragma unroll
    for (int cam = 0; cam < NCAM; ++cam)
        s += osca[(size_t)cam * QN * CE + i] * hitf[cam * QN + q];
    slots[i] = s / cnt[q];
}

// ===========================================================================
extern "C" void kernel_launch(void* const* d_in, const int* in_sizes, int n_in,
                              void* d_out, int out_size, void* d_ws, size_t ws_size,
                              hipStream_t stream)
{
    (void)in_sizes; (void)n_in; (void)out_size; (void)ws_size;

    const float* query   = (const float*)d_in[0];
    const float* value   = (const float*)d_in[1];   // (6,19560,1,256) == (117360,256)
    const float* bev_pos = (const float*)d_in[2];
    const float* ref2d   = (const float*)d_in[3];
    const float* refimg  = (const float*)d_in[4];
    const float* prevb   = (const float*)d_in[5];   // (2,QN,256)

    const float* tsa_vw  = (const float*)d_in[6];
    const float* tsa_vb  = (const float*)d_in[7];
    const float* tsa_ow  = (const float*)d_in[8];
    const float* tsa_ob  = (const float*)d_in[9];
    const float* tsa_aww = (const float*)d_in[10];
    const float* tsa_awb = (const float*)d_in[11];
    const float* tsa_pw  = (const float*)d_in[12];
    const float* tsa_pb  = (const float*)d_in[13];
    const float* sca_vw  = (const float*)d_in[14];
    const float* sca_vb  = (const float*)d_in[15];
    const float* sca_ow  = (const float*)d_in[16];
    const float* sca_ob  = (const float*)d_in[17];
    const float* sca_aww = (const float*)d_in[18];
    const float* sca_awb = (const float*)d_in[19];
    const float* sca_pw  = (const float*)d_in[20];
    const float* sca_pb  = (const float*)d_in[21];
    const float* ln0_g   = (const float*)d_in[22];
    const float* ln0_b   = (const float*)d_in[23];
    const float* ln1_g   = (const float*)d_in[24];
    const float* ln1_b   = (const float*)d_in[25];
    const float* ln2_g   = (const float*)d_in[26];
    const float* ln2_b   = (const float*)d_in[27];
    const float* ffn_w1  = (const float*)d_in[28];
    const float* ffn_b1  = (const float*)d_in[29];
    const float* ffn_w2  = (const float*)d_in[30];
    const float* ffn_b2  = (const float*)d_in[31];
    const unsigned char* bmask = (const unsigned char*)d_in[34];
    float* outp = (float*)d_out;

    // ----------------- workspace carve-up (floats) -----------------
    float* ws = (float*)d_ws;
    size_t off = 0;
    auto alloc = [&](size_t n) { float* p = ws + off; off += n; return p; };
    float* qbuf    = alloc((size_t)QN * CE);        // q = query + bev_pos
    float* q2      = alloc((size_t)QN * 512);       // concat(prev_bev[0], q)
    float* tsa_v   = alloc((size_t)2 * QN * CE);
    float* tsa_off = alloc((size_t)QN * 128);
    float* tsa_aw  = alloc((size_t)QN * 64);
    float* tsa_ms  = alloc((size_t)2 * QN * CE);
    float* omean   = alloc((size_t)QN * CE);
    float* tmp0    = alloc((size_t)QN * CE);        // pre-LN scratch (reused)
    float* qln0    = alloc((size_t)QN * CE);
    float* sca_off = alloc((size_t)QN * 512);
    float* sca_aw  = alloc((size_t)QN * 256);
    float* o_sca   = alloc((size_t)NCAM * QN * CE);
    float* hitf    = alloc((size_t)NCAM * QN);
    float* cnt     = alloc((size_t)QN);
    float* slots   = alloc((size_t)QN * CE);
    float* qln1    = alloc((size_t)QN * CE);
    float* ffnh    = alloc((size_t)QN * FFD);
    float* sca_v   = alloc((size_t)NCAM * LTOT * CE);   // big one last (~120 MB)

    auto gemm = [&](const float* A, const float* W, const float* B, const float* R,
                    float* O, int M, int N, int K, int relu) {
        int tiles = ((M + 15) / 16) * ((N + 63) / 64);
        gemm_wmma_bias<<<(tiles + 3) / 4, 128, 0, stream>>>(A, W, B, R, O,
                                                            M, N, K, relu);
    };
    const int EW = 256;

    // ---------------- TSA ----------------
    int nQC = QN * CE;
    add_kernel<<<(nQC + EW - 1) / EW, EW, 0, stream>>>(query, bev_pos, qbuf, nQC);
    int nQ2 = QN * 512;
    concat_kernel<<<(nQ2 + EW - 1) / EW, EW, 0, stream>>>(prevb, qbuf, q2, nQ2);

    gemm(prevb, tsa_vw,  tsa_vb,  nullptr, tsa_v,   2 * QN, CE,  CE,  0);  // value proj
    gemm(q2,    tsa_ow,  tsa_ob,  nullptr, tsa_off, QN,     128, 512, 0);  // offsets
    gemm(q2,    tsa_aww, tsa_awb, nullptr, tsa_aw,  QN,     64,  512, 0);  // attn logits
    softmax4_kernel<<<(QN * 16 + EW - 1) / EW, EW, 0, stream>>>(tsa_aw, QN * 16);

    tsa_msda_kernel<<<(2 * QN * NH + 7) / 8, 256, 0, stream>>>(tsa_v, tsa_off,
                                                               tsa_aw, ref2d, tsa_ms);
    mean2_kernel<<<(nQC + EW - 1) / EW, EW, 0, stream>>>(tsa_ms, omean, nQC);

    gemm(omean, tsa_pw, tsa_pb, query, tmp0, QN, CE, CE, 0);               // out proj + resid
    layernorm_kernel<<<QN, 32, 0, stream>>>(tmp0, ln0_g, ln0_b, qln0);

    // ---------------- SCA ----------------
    {   // big value projection: TDM-staged weights, double-buffered in LDS
        int Mv = NCAM * LTOT;                           // 117360 rows
        int blocks = (Mv + 31) / 32;                    // 32 rows per block
        size_t smem = (size_t)2 * TBUF * sizeof(float); // ~68 KB
        gemm_wmma_tdm<<<blocks, 256, smem, stream>>>(value, sca_vw, sca_vb,
                                                     sca_v, Mv);
    }
    // qc is the same query broadcast over cameras -> compute projections once
    gemm(qln0, sca_ow,  sca_ob,  nullptr, sca_off, QN, 512, CE, 0);
    gemm(qln0, sca_aww, sca_awb, nullptr, sca_aw,  QN, 256, CE, 0);
    softmax32_kernel<<<(QN * NH + EW - 1) / EW, EW, 0, stream>>>(sca_aw, QN * NH);

    hit_kernel<<<(QN + EW - 1) / EW, EW, 0, stream>>>(bmask, hitf, cnt);
    sca_msda_kernel<<<(NCAM * QN * NH + 7) / 8, 256, 0, stream>>>(sca_v, sca_off,
                                                                  sca_aw, refimg, o_sca);
    slots_kernel<<<(nQC + EW - 1) / EW, EW, 0, stream>>>(o_sca, hitf, cnt, slots, nQC);

    gemm(slots, sca_pw, sca_pb, qln0, tmp0, QN, CE, CE, 0);                // out proj + resid
    layernorm_kernel<<<QN, 32, 0, stream>>>(tmp0, ln1_g, ln1_b, qln1);

    // ---------------- FFN ----------------
    gemm(qln1, ffn_w1, ffn_b1, nullptr, ffnh, QN, FFD, CE, 1);             // + ReLU
    gemm(ffnh, ffn_w2, ffn_b2, qln1,    tmp0, QN, CE, FFD, 0);             // + resid
    layernorm_kernel<<<QN, 32, 0, stream>>>(tmp0, ln2_g, ln2_b, outp);
}